// BipartiteSAGEExtended_33603824124606
// MI455X (gfx1250) — compile-verified
//
#include <hip/hip_runtime.h>

// ---------------------------------------------------------------------------
// BipartiteSAGE on gfx1250: encoders + 2 SAGE layers.
// Dense math via V_WMMA_F32_16X16X4_F32 (fp32-exact), edge aggregation via
// coalesced gather + global_atomic_add_f32 scatter.
// Workspace layout (floats): h[N*128] | s[N*128] | cnt[N]   (~103 MB)
// ---------------------------------------------------------------------------

typedef __attribute__((ext_vector_type(2))) float v2f;
typedef __attribute__((ext_vector_type(8))) float v8f;

#define LDK 260  // LDS row stride (floats). 260 % 64 == 4 -> conflict-free
                 // column reads for 16 lanes, even -> 8B-aligned pair loads.

__device__ __forceinline__ v8f wmma4(v2f a, v2f b, v8f c) {
  // D(16x16,f32) = A(16x4,f32) * B(4x16,f32) + C
  return __builtin_amdgcn_wmma_f32_16x16x4_f32(false, a, false, b,
                                               (short)0, c, false, false);
}

// Accumulate A_lds[16 x K] (row stride LDK, starting column aoff) times
// row-major B[K x 128] (columns col0..col0+15) into c.
__device__ __forceinline__ void gemm_acc(const float* As, int aoff,
                                         const float* __restrict__ B,
                                         int K, int col0, int lane, v8f& c) {
  const int hl = lane >> 4;   // wave half selects K sub-pair
  const int m  = lane & 15;   // A row
  const int n  = lane & 15;   // B column within tile
  #pragma unroll 4
  for (int k0 = 0; k0 < K; k0 += 4) {
    const int ka = k0 + 2 * hl;
    v2f a, b;
    a.x = As[m * LDK + aoff + ka];
    a.y = As[m * LDK + aoff + ka + 1];
    b.x = B[ka * 128 + col0 + n];
    b.y = B[(ka + 1) * 128 + col0 + n];
    c = wmma4(a, b, c);
  }
}

__device__ __forceinline__ void store_tile(float* __restrict__ out, int row0,
                                           int nrows, int col0, int lane,
                                           v8f c, bool do_relu) {
  const int hl = lane >> 4;
  const int n  = lane & 15;
  #pragma unroll
  for (int v = 0; v < 8; ++v) {
    int row = row0 + v + 8 * hl;          // C layout: VGPR v -> M = v + 8*half
    float val = c[v];
    if (do_relu) val = fmaxf(val, 0.0f);
    if (row < nrows) out[(size_t)row * 128 + col0 + n] = val;
  }
}

// ------------------------- encoder kernels ---------------------------------

__global__ void __launch_bounds__(256) encode_pol_kernel(
    const float* __restrict__ x, const int* __restrict__ sidx,
    const float* __restrict__ semb, const float* __restrict__ W,
    const float* __restrict__ bvec, float* __restrict__ out, int nrows) {
  __shared__ float As[16 * LDK];
  const int tid  = threadIdx.x;
  const int row0 = blockIdx.x * 16;
  // Stage concat([x(64) | state_emb(8)]) -> A_lds[16 x 72]
  for (int idx = tid; idx < 16 * 72; idx += 256) {
    int r = idx / 72, cc = idx % 72;
    int row = row0 + r;
    float v = 0.0f;
    if (row < nrows) {
      if (cc < 64) v = x[(size_t)row * 64 + cc];
      else         v = semb[sidx[row] * 8 + (cc - 64)];
    }
    As[r * LDK + cc] = v;
  }
  __syncthreads();
  const int lane = tid & 31;
  const int col0 = (tid >> 5) * 16;
  v8f c;
  float bias = bvec[col0 + (lane & 15)];
  #pragma unroll
  for (int v = 0; v < 8; ++v) c[v] = bias;
  gemm_acc(As, 0, W, 72, col0, lane, c);
  store_tile(out, row0, nrows, col0, lane, c, true);
}

__global__ void __launch_bounds__(256) encode_comp_kernel(
    const float* __restrict__ x, const int* __restrict__ secidx,
    const int* __restrict__ indidx, const float* __restrict__ secemb,
    const float* __restrict__ indemb, const float* __restrict__ W,
    const float* __restrict__ bvec, float* __restrict__ out, int nrows) {
  __shared__ float As[16 * LDK];
  const int tid  = threadIdx.x;
  const int row0 = blockIdx.x * 16;
  // Stage concat([x(96) | sector_emb(8) | ind_emb(8)]) -> A_lds[16 x 112]
  for (int idx = tid; idx < 16 * 112; idx += 256) {
    int r = idx / 112, cc = idx % 112;
    int row = row0 + r;
    float v = 0.0f;
    if (row < nrows) {
      if (cc < 96)       v = x[(size_t)row * 96 + cc];
      else if (cc < 104) v = secemb[secidx[row] * 8 + (cc - 96)];
      else               v = indemb[indidx[row] * 8 + (cc - 104)];
    }
    As[r * LDK + cc] = v;
  }
  __syncthreads();
  const int lane = tid & 31;
  const int col0 = (tid >> 5) * 16;
  v8f c;
  float bias = bvec[col0 + (lane & 15)];
  #pragma unroll
  for (int v = 0; v < 8; ++v) c[v] = bias;
  gemm_acc(As, 0, W, 112, col0, lane, c);
  store_tile(out, row0, nrows, col0, lane, c, true);
}

// --------------------- aggregation (mean over in-edges) --------------------

__global__ void zero_kernel(float* __restrict__ p, size_t n) {
  size_t i      = (size_t)blockIdx.x * blockDim.x + threadIdx.x;
  size_t stride = (size_t)gridDim.x * blockDim.x;
  for (; i < n; i += stride) p[i] = 0.0f;
}

// One wave per edge: lane l moves h[src][4l..4l+3] into s[dst] via f32 atomics.
__global__ void __launch_bounds__(256) scatter_kernel(
    const float* __restrict__ h, const int* __restrict__ src,
    const int* __restrict__ dst, float* __restrict__ s,
    float* __restrict__ cnt, int E) {
  const int lane = threadIdx.x & 31;
  const int e    = blockIdx.x * 8 + (threadIdx.x >> 5);
  if (e >= E) return;
  const int sv = src[e];
  const int dv = dst[e];
  const float4 v = *reinterpret_cast<const float4*>(h + (size_t)sv * 128 + lane * 4);
  float* p = s + (size_t)dv * 128 + lane * 4;
  atomicAdd(p + 0, v.x);
  atomicAdd(p + 1, v.y);
  atomicAdd(p + 2, v.z);
  atomicAdd(p + 3, v.w);
  if (lane == 0) atomicAdd(cnt + dv, 1.0f);
}

// ------------------- fused SAGE layer: relu?(m*Wl + bl + h*Wr) -------------

__global__ void __launch_bounds__(256) sage_kernel(
    const float* __restrict__ h, const float* __restrict__ s,
    const float* __restrict__ cnt, const float* __restrict__ Wl,
    const float* __restrict__ bl, const float* __restrict__ Wr,
    float* __restrict__ out, int nrows, int do_relu) {
  __shared__ float As[16 * LDK];  // cols [0,128) = mean, [128,256) = h
  const int tid  = threadIdx.x;
  const int row0 = blockIdx.x * 16;
  for (int idx = tid; idx < 16 * 256; idx += 256) {
    int r = idx >> 8, cc = idx & 255;
    int row = row0 + r;
    float v = 0.0f;
    if (row < nrows) {
      if (cc < 128) {
        v = s[(size_t)row * 128 + cc] / fmaxf(cnt[row], 1.0f);  // fused mean
      } else {
        v = h[(size_t)row * 128 + (cc - 128)];
      }
    }
    As[r * LDK + cc] = v;
  }
  __syncthreads();  // all global reads of h done -> in-place update of h is safe
  const int lane = tid & 31;
  const int col0 = (tid >> 5) * 16;
  v8f c;
  float bias = bl[col0 + (lane & 15)];
  #pragma unroll
  for (int v = 0; v < 8; ++v) c[v] = bias;
  gemm_acc(As, 0,   Wl, 128, col0, lane, c);   // m @ Wl
  gemm_acc(As, 128, Wr, 128, col0, lane, c);   // h @ Wr
  store_tile(out, row0, nrows, col0, lane, c, do_relu != 0);
}

// ---------------------------------------------------------------------------

extern "C" void kernel_launch(void* const* d_in, const int* in_sizes, int n_in,
                              void* d_out, int out_size, void* d_ws, size_t ws_size,
                              hipStream_t stream) {
  (void)n_in; (void)out_size; (void)ws_size;
  const float* x_pol   = (const float*)d_in[0];
  const int*   pol_sid = (const int*)  d_in[1];
  const float* x_comp  = (const float*)d_in[2];
  const int*   c_sec   = (const int*)  d_in[3];
  const int*   c_ind   = (const int*)  d_in[4];
  const int*   edges   = (const int*)  d_in[5];
  const float* s_emb   = (const float*)d_in[6];
  const float* sec_emb = (const float*)d_in[7];
  const float* ind_emb = (const float*)d_in[8];
  const float* W_pol   = (const float*)d_in[9];
  const float* b_pol   = (const float*)d_in[10];
  const float* W_comp  = (const float*)d_in[11];
  const float* b_comp  = (const float*)d_in[12];
  const float* Wl1     = (const float*)d_in[13];
  const float* bl1     = (const float*)d_in[14];
  const float* Wr1     = (const float*)d_in[15];
  const float* Wl2     = (const float*)d_in[16];
  const float* bl2     = (const float*)d_in[17];
  const float* Wr2     = (const float*)d_in[18];

  const int NP = in_sizes[0] / 64;
  const int NC = in_sizes[2] / 96;
  const int E  = in_sizes[5] / 2;
  const int N  = NP + NC;
  const int* src = edges;       // edge_index[0]
  const int* dst = edges + E;   // edge_index[1]

  float* h    = (float*)d_ws;
  float* s    = h + (size_t)N * 128;
  float* cnt  = s + (size_t)N * 128;
  float* outp = (float*)d_out;

  const dim3 blk(256);
  const size_t zn = (size_t)N * 128 + N;   // s and cnt are contiguous
  int zgrid = (int)((zn + 255) / 256);
  if (zgrid > 65535) zgrid = 65535;        // grid-stride covers remainder

  // node encoders -> h
  encode_pol_kernel<<<dim3((NP + 15) / 16), blk, 0, stream>>>(
      x_pol, pol_sid, s_emb, W_pol, b_pol, h, NP);
  encode_comp_kernel<<<dim3((NC + 15) / 16), blk, 0, stream>>>(
      x_comp, c_sec, c_ind, sec_emb, ind_emb, W_comp, b_comp,
      h + (size_t)NP * 128, NC);

  // SAGE layer 1 (ReLU), h updated in place
  zero_kernel<<<dim3(zgrid), blk, 0, stream>>>(s, zn);
  scatter_kernel<<<dim3((E + 7) / 8), blk, 0, stream>>>(h, src, dst, s, cnt, E);
  sage_kernel<<<dim3((N + 15) / 16), blk, 0, stream>>>(
      h, s, cnt, Wl1, bl1, Wr1, h, N, 1);

  // SAGE layer 2 (no ReLU) -> d_out ( = [h_pol ; h_comp] )
  zero_kernel<<<dim3(zgrid), blk, 0, stream>>>(s, zn);
  scatter_kernel<<<dim3((E + 7) / 8), blk, 0, stream>>>(h, src, dst, s, cnt, E);
  sage_kernel<<<dim3((N + 15) / 16), blk, 0, stream>>>(
      h, s, cnt, Wl2, bl2, Wr2, outp, N, 0);
}